// RelGraphConv_13331578487268
// MI455X (gfx1250) — compile-verified
//
#include <hip/hip_runtime.h>

// RelGraphConv (basis decomposition) for MI455X / gfx1250.
// Pipeline: basis compose -> relation-chunked WMMA fp32 GEMM (h_chunk stays in
// 192MB L2) -> gather*norm + atomic scatter into L2-resident agg -> fused
// self-loop WMMA GEMM + bias + relu epilogue.

typedef __attribute__((ext_vector_type(2))) float v2f;
typedef __attribute__((ext_vector_type(4))) float v4f;
typedef __attribute__((ext_vector_type(8))) float v8f;

#define N_NODES 20000
#define N_EDGES 640000
#define DIM     128      // IN == OUT
#define N_REL   64
#define N_BASIS 16
#define RC      8        // relations per chunk (h_chunk = 82MB, fits L2)
#define N_CHUNK (N_REL / RC)
#define MBLK    128      // rows per block (8 waves x 16)
#define NBLK    64       // cols per block (4 wmma tiles per wave)
#define SA      132      // LDS row stride (floats), padded, even for b64 loads
#define SB      132

// ---------------------------------------------------------------------------
// Shared GEMM core: C[128x64] += A[128x128] * B[128x64] using
// V_WMMA_F32_16X16X4_F32. A = x rows (zero-padded past N_NODES),
// B = column panel with row stride DIM. Result left in acc[4] (v8f each).
// ---------------------------------------------------------------------------
__device__ __forceinline__ void gemm_core(const float* __restrict__ x,
                                          const float* __restrict__ Bmat,
                                          int rowBase,
                                          float* ldsA, float* ldsB,
                                          v8f acc[4]) {
  const int tid = threadIdx.x;

  // Stage A panel: 128 rows x 128 floats, coalesced float4 loads.
  for (int i = tid; i < MBLK * (DIM / 4); i += 256) {
    int row = i >> 5;              // 32 float4 per row
    int c4  = (i & 31) << 2;
    int m   = rowBase + row;
    v4f v = {};
    if (m < N_NODES) v = *(const v4f*)(x + (size_t)m * DIM + c4);
    *(v4f*)(ldsA + row * SA + c4) = v;   // (132*row + 4c) % 4 == 0 -> 16B aligned
  }
  // Stage B panel transposed: ldsB[n*SB + k] = B[k*DIM + n] so that the
  // per-lane K-pair {B[k][n], B[k+1][n]} is a contiguous ds_load_b64.
  for (int i = tid; i < DIM * NBLK; i += 256) {
    int k = i >> 6;
    int n = i & 63;
    ldsB[n * SB + k] = Bmat[(size_t)k * DIM + n];
  }
  __syncthreads();

  const int lane = tid & 31;
  const int wave = tid >> 5;                 // 8 waves, 16 rows each
  const int lm   = lane & 15;                // M (for A) / N (for B) index
  const int lk   = (lane >> 4) << 1;         // lanes 16-31 carry K+2
  const float* aRow  = ldsA + (wave * 16 + lm) * SA + lk;
  const float* bCol0 = ldsB + lm * SB + lk;

  v8f z = {};
  acc[0] = z; acc[1] = z; acc[2] = z; acc[3] = z;

#pragma unroll 8
  for (int kk = 0; kk < DIM; kk += 4) {
    v2f a = *(const v2f*)(aRow + kk);        // {A[m][kk+lk], A[m][kk+lk+1]}
#pragma unroll
    for (int t = 0; t < 4; ++t) {
      v2f b = *(const v2f*)(bCol0 + t * 16 * SB + kk);
      acc[t] = __builtin_amdgcn_wmma_f32_16x16x4_f32(
          false, a, false, b, (short)0, acc[t], false, false);
    }
  }
}

// ---------------------------------------------------------------------------
// h_chunk[m][rc*128 + n] = sum_i x[m][i] * W_rel[r0+rc][i][n]
// grid = (ceil(N/128), RC*2): blockIdx.y -> (relation-in-chunk, 64-col half)
// ---------------------------------------------------------------------------
__global__ void __launch_bounds__(256)
gemm_chunk_kernel(const float* __restrict__ x, const float* __restrict__ Wrel,
                  float* __restrict__ hchunk, int r0) {
  __shared__ float ldsA[MBLK * SA];
  __shared__ float ldsB[NBLK * SB];
  const int rowBase = blockIdx.x * MBLK;
  const int rc = blockIdx.y >> 1;
  const int n0 = (blockIdx.y & 1) * NBLK;
  const float* Bmat = Wrel + ((size_t)(r0 + rc)) * DIM * DIM + n0;

  v8f acc[4];
  gemm_core(x, Bmat, rowBase, ldsA, ldsB, acc);

  const int lane = threadIdx.x & 31;
  const int wave = threadIdx.x >> 5;
  const int mlo  = rowBase + wave * 16 + ((lane >> 4) << 3);  // M = v + 8*hi
  const int ncol = rc * DIM + n0 + (lane & 15);
#pragma unroll
  for (int t = 0; t < 4; ++t)
#pragma unroll
    for (int v = 0; v < 8; ++v) {
      int m = mlo + v;
      if (m < N_NODES)
        hchunk[(size_t)m * (RC * DIM) + ncol + t * 16] = acc[t][v];
    }
}

// out = relu(acc + agg + bias), acc = x @ loop_weight. grid = (ceil(N/128), 2)
__global__ void __launch_bounds__(256)
final_kernel(const float* __restrict__ x, const float* __restrict__ loopW,
             const float* __restrict__ agg, const float* __restrict__ bias,
             float* __restrict__ out) {
  __shared__ float ldsA[MBLK * SA];
  __shared__ float ldsB[NBLK * SB];
  const int rowBase = blockIdx.x * MBLK;
  const int n0 = blockIdx.y * NBLK;

  v8f acc[4];
  gemm_core(x, loopW + n0, rowBase, ldsA, ldsB, acc);

  const int lane = threadIdx.x & 31;
  const int wave = threadIdx.x >> 5;
  const int mlo  = rowBase + wave * 16 + ((lane >> 4) << 3);
  const int nc   = n0 + (lane & 15);
#pragma unroll
  for (int t = 0; t < 4; ++t)
#pragma unroll
    for (int v = 0; v < 8; ++v) {
      int m = mlo + v;
      if (m < N_NODES) {
        int col = nc + t * 16;
        float r = acc[t][v] + agg[(size_t)m * DIM + col] + bias[col];
        out[(size_t)m * DIM + col] = r > 0.f ? r : 0.f;
      }
    }
}

// W_rel[r][i][o] = sum_b w_comp[r][b] * weight[b][i][o]
__global__ void __launch_bounds__(256)
basis_kernel(const float* __restrict__ wc, const float* __restrict__ w,
             float* __restrict__ Wrel) {
  int idx = blockIdx.x * 256 + threadIdx.x;        // R * 128 * 128 total
  int r  = idx >> 14;
  int io = idx & 16383;
  float s = 0.f;
#pragma unroll
  for (int b = 0; b < N_BASIS; ++b)
    s += wc[r * N_BASIS + b] * w[(size_t)b * DIM * DIM + io];
  Wrel[idx] = s;
}

// agg[dst[e]] += norm[e] * h_chunk[src[e]][etype[e]-r0]; 32 lanes/edge, float4
__global__ void __launch_bounds__(256)
scatter_kernel(const float* __restrict__ hchunk, const int* __restrict__ src,
               const int* __restrict__ dst, const int* __restrict__ ety,
               const float* __restrict__ norm, float* __restrict__ agg, int r0) {
  int e = blockIdx.x * 8 + (threadIdx.x >> 5);
  if (e >= N_EDGES) return;
  int r = ety[e] - r0;
  if ((unsigned)r >= (unsigned)RC) return;         // not this chunk
  int lane = threadIdx.x & 31;
  float nv = norm[e];
  const v4f* hrow = (const v4f*)(hchunk + ((size_t)src[e] * RC + r) * DIM);
  v4f v = hrow[lane];                              // 32 lanes x 16B = 128 floats
  float* ap = agg + (size_t)dst[e] * DIM + lane * 4;
  atomicAdd(ap + 0, v.x * nv);
  atomicAdd(ap + 1, v.y * nv);
  atomicAdd(ap + 2, v.z * nv);
  atomicAdd(ap + 3, v.w * nv);
}

__global__ void zero_kernel(float* __restrict__ p, int n) {
  int i = (blockIdx.x * 256 + threadIdx.x) * 4;
  v4f z = {};
  if (i < n) *(v4f*)(p + i) = z;
}

extern "C" void kernel_launch(void* const* d_in, const int* in_sizes, int n_in,
                              void* d_out, int out_size, void* d_ws, size_t ws_size,
                              hipStream_t stream) {
  const float* x     = (const float*)d_in[0];
  const float* weight= (const float*)d_in[1];
  const float* wcomp = (const float*)d_in[2];
  const float* hbias = (const float*)d_in[3];
  const float* loopW = (const float*)d_in[4];
  const float* norm  = (const float*)d_in[5];
  const int*   src   = (const int*)d_in[6];
  const int*   dst   = (const int*)d_in[7];
  const int*   ety   = (const int*)d_in[8];
  float* out = (float*)d_out;

  // Workspace layout (floats): Wrel (4MB) | agg (10.24MB) | h_chunk (81.92MB)
  float* ws     = (float*)d_ws;
  float* Wrel   = ws;
  float* agg    = Wrel + (size_t)N_REL * DIM * DIM;
  float* hchunk = agg + (size_t)N_NODES * DIM;

  zero_kernel<<<(N_NODES * DIM / 4 + 255) / 256, 256, 0, stream>>>(agg, N_NODES * DIM);
  basis_kernel<<<(N_REL * DIM * DIM) / 256, 256, 0, stream>>>(wcomp, weight, Wrel);

  dim3 gg((N_NODES + MBLK - 1) / MBLK, RC * 2);
  for (int c = 0; c < N_CHUNK; ++c) {
    gemm_chunk_kernel<<<gg, 256, 0, stream>>>(x, Wrel, hchunk, c * RC);
    scatter_kernel<<<(N_EDGES + 7) / 8, 256, 0, stream>>>(hchunk, src, dst, ety,
                                                          norm, agg, c * RC);
  }
  final_kernel<<<dim3((N_NODES + MBLK - 1) / MBLK, DIM / NBLK), 256, 0, stream>>>(
      x, loopW, agg, hbias, out);
}